// ClassLogitContrastiveLoss_27865747816628
// MI455X (gfx1250) — compile-verified
//
#include <hip/hip_runtime.h>

typedef float v2f __attribute__((ext_vector_type(2)));
typedef float v8f __attribute__((ext_vector_type(8)));

namespace {
constexpr int kBL      = 256;   // B*L batches
constexpr int kN       = 512;   // points per batch
constexpr int kJ       = 128;   // logit dim
constexpr int kThreads = 256;   // 8 wave32
constexpr int kWaves   = kThreads / 32;
constexpr int kTiles   = kN / 16;   // 32 tiles per matrix dim
}

__global__ __launch_bounds__(kThreads)
void cl_loss_main(const float* __restrict__ ypred,
                  const float* __restrict__ xyz,
                  float* __restrict__ block_sums)
{
    __shared__ float xs[kN], ys[kN], zs[kN], sqr[kN];
    __shared__ int   maxidx[kN], minidx[kN];
    __shared__ float wsum[kWaves];

    const int bl   = blockIdx.x;
    const int tid  = threadIdx.x;
    const int wave = tid >> 5;
    const int lane = tid & 31;
    const int half = lane >> 4;    // 0: lanes 0-15, 1: lanes 16-31
    const int lm   = lane & 15;

    // ---------- Phase A: stage this batch's xyz in LDS, compute |x|^2 ----------
    const float* xb = xyz + (size_t)bl * kN * 3;
    for (int n = tid; n < kN; n += kThreads) {
        float x = xb[n * 3 + 0];
        float y = xb[n * 3 + 1];
        float z = xb[n * 3 + 2];
        xs[n] = x; ys[n] = y; zs[n] = z;
        sqr[n] = x * x + y * y + z * z;
    }
    __syncthreads();

    // ---------- Phase B: WMMA Gram tiles + fused per-row argmax / masked argmin --------
    // d2(n,m) = sq[n] + sq[m] - 2 * <x_n, x_m>   (the reference's mm trick)
    for (int ti = wave; ti < kTiles; ti += kWaves) {
        // A operand (16x4 f32): lanes 0-15 hold K0=x, K1=y; lanes 16-31 hold K2=z, K3=0
        const int rowA = ti * 16 + lm;
        v2f A;
        A.x = half ? zs[rowA] : xs[rowA];
        A.y = half ? 0.0f     : ys[rowA];

        float rsq[8]; int rix[8];
        float mxv[8]; int mxi[8];
        float mnv[8]; int mni[8];
        #pragma unroll
        for (int v = 0; v < 8; ++v) {
            rix[v] = ti * 16 + v + half * 8;  // D layout: VGPR v <-> row v (+8 in hi half)
            rsq[v] = sqr[rix[v]];
            mxv[v] = -1.0f;   mxi[v] = 0;
            mnv[v] = 3.0e38f; mni[v] = 0;
        }

        for (int tj = 0; tj < kTiles; ++tj) {
            const int m = tj * 16 + lm;      // this lane's column in the tile
            // B operand (4x16 f32), transpose-mirror of A layout
            v2f Bv;
            Bv.x = half ? zs[m] : xs[m];
            Bv.y = half ? 0.0f : ys[m];
            v8f C = {};
            v8f G = __builtin_amdgcn_wmma_f32_16x16x4_f32(
                        false, A, false, Bv, (short)0, C, false, false);
            #pragma unroll
            for (int v = 0; v < 8; ++v) {
                float d2 = rsq[v] + sqr[m] - 2.0f * G[v];
                d2 = fmaxf(d2, 0.0f);
                const bool diag = (m == rix[v]);
                // farthest: diagonal distance forced to 0 (eye mask in reference)
                float dmax = diag ? 0.0f : d2;
                if (dmax > mxv[v]) { mxv[v] = dmax; mxi[v] = m; }
                // nearest: mask self / exact duplicates (dist==0 -> huge)
                float dmin = (diag || d2 == 0.0f) ? 3.0e38f : d2;
                if (dmin < mnv[v]) { mnv[v] = dmin; mni[v] = m; }
            }
        }

        // Butterfly reduce across the 16 lanes sharing each row (xor<=8 stays in-half).
        // Tie-break toward the smaller index to match jnp.argmax/argmin semantics.
        #pragma unroll
        for (int off = 1; off <= 8; off <<= 1) {
            #pragma unroll
            for (int v = 0; v < 8; ++v) {
                float ov = __shfl_xor(mxv[v], off, 32);
                int   oi = __shfl_xor(mxi[v], off, 32);
                if (ov > mxv[v] || (ov == mxv[v] && oi < mxi[v])) { mxv[v] = ov; mxi[v] = oi; }
                float pv = __shfl_xor(mnv[v], off, 32);
                int   pi = __shfl_xor(mni[v], off, 32);
                if (pv < mnv[v] || (pv == mnv[v] && pi < mni[v])) { mnv[v] = pv; mni[v] = pi; }
            }
        }
        if (lm == 0) {
            #pragma unroll
            for (int v = 0; v < 8; ++v) {
                maxidx[rix[v]] = mxi[v];
                minidx[rix[v]] = mni[v];
            }
        }
    }
    __syncthreads();

    // ---------- Phase C: sim at argmax/argmin only ----------
    // yp[n][j] = ypred[bl][j][n]; contribution = <yp[n], yp[mx] - yp[mn]>
    const float* yb = ypred + (size_t)bl * kJ * kN;
    float acc = 0.0f;
    for (int chunk = wave; chunk < kN / 32; chunk += kWaves) {
        const int n  = chunk * 32 + lane;
        const int mi = maxidx[n];
        const int ni = minidx[n];
        float a = 0.0f;
        for (int j = 0; j < kJ; ++j) {
            const float* row = yb + j * kN;      // 2KB row, L2/L0 resident
            a = fmaf(row[n], row[mi] - row[ni], a);
        }
        acc += a;
    }
    #pragma unroll
    for (int off = 16; off >= 1; off >>= 1)
        acc += __shfl_xor(acc, off, 32);
    if (lane == 0) wsum[wave] = acc;
    __syncthreads();
    if (tid == 0) {
        float t = 0.0f;
        #pragma unroll
        for (int w = 0; w < kWaves; ++w) t += wsum[w];
        block_sums[bl] = t;   // deterministic per-block partial
    }
}

__global__ __launch_bounds__(kThreads)
void cl_loss_finalize(const float* __restrict__ block_sums,
                      float* __restrict__ out)
{
    __shared__ float s[kWaves];
    const int tid = threadIdx.x;
    float v = block_sums[tid];            // 256 partials, 256 threads
    #pragma unroll
    for (int off = 16; off >= 1; off >>= 1)
        v += __shfl_xor(v, off, 32);
    if ((tid & 31) == 0) s[tid >> 5] = v;
    __syncthreads();
    if (tid == 0) {
        float t = 0.0f;
        #pragma unroll
        for (int w = 0; w < kWaves; ++w) t += s[w];
        out[0] = t / (float)(kBL * kN);   // mean over all (batch, point)
    }
}

extern "C" void kernel_launch(void* const* d_in, const int* in_sizes, int n_in,
                              void* d_out, int out_size, void* d_ws, size_t ws_size,
                              hipStream_t stream) {
    const float* ypred = (const float*)d_in[0];   // [8,32,128,512] f32
    const float* xyz   = (const float*)d_in[1];   // [8,32,512,3]  f32
    float* block_sums  = (float*)d_ws;            // 256 floats of scratch

    cl_loss_main<<<kBL, kThreads, 0, stream>>>(ypred, xyz, block_sums);
    cl_loss_finalize<<<1, kThreads, 0, stream>>>(block_sums, (float*)d_out);
}